// LightSB_73718818668756
// MI455X (gfx1250) — compile-verified
//
#include <hip/hip_runtime.h>
#include <hip/hip_bf16.h>
#include <math.h>

// ---------------------------------------------------------------------------
// LightSB sampling kernel for gfx1250 (MI455X).
//
// logits[b,k] = sum_d ( x[b,d]*r[k,d] + 0.5*x[b,d]^2*exp(Slog[k,d]) ) + la[k]
// idx[b]      = argmax_k ( logits[b,k] + gumbel(u[b,k]) )
// out[b,d]    = r[idx,d] + S[idx,d]*x[b,d] + sqrt(S[idx,d])*noise[b,d]
//
// fp32 WMMA (V_WMMA_F32_16X16X4_F32), A = [x | 0.5 x^2], W^T = [r ; S],
// K-dim = 128.  ~512 MB mandatory HBM traffic -> ~22 us roofline floor.
// This revision: single base-VGPR + immediate-offset LDS addressing (kills
// the >256-VGPR allocation), one-step software pipeline on fragment loads,
// hardware log/sqrt to cut VALU below the bandwidth floor, u read directly
// from global in C-matrix layout.
// ---------------------------------------------------------------------------

typedef __attribute__((ext_vector_type(2))) float v2f;
typedef __attribute__((ext_vector_type(8))) float v8f;

#define WAVES     8
#define BLOCK     (WAVES * 32)
#define DD        64
#define KK        64
#define XSTRIDE   68          // 16-row tile, padded row stride (floats)
#define WTSTRIDE  132         // Wt row stride (floats), 128 cols + pad

__launch_bounds__(BLOCK)
__global__ void lightsb_wmma_kernel(const float* __restrict__ x,
                                    const float* __restrict__ log_alpha_raw,
                                    const float* __restrict__ r,
                                    const float* __restrict__ S_log_diag,
                                    const float* __restrict__ u,
                                    const float* __restrict__ noise,
                                    float* __restrict__ out,
                                    int numTiles)
{
    // Wt[k][j]: j<64 -> r[k][j], j>=64 -> exp(S_log_diag[k][j-64])
    __shared__ float Wt[KK * WTSTRIDE];
    __shared__ float la[KK];
    __shared__ float xbuf[WAVES][16 * XSTRIDE];
    __shared__ float lbuf[WAVES][16 * XSTRIDE];   // logits+gumbel for argmax
    __shared__ int   idxbuf[WAVES][16];

    const int tid = threadIdx.x;

    // ---- cooperative fused-weight build (once per block) ----
    for (int i = tid; i < KK * 128; i += BLOCK) {
        int k = i >> 7, j = i & 127;
        float v = (j < 64) ? r[k * 64 + j] : expf(S_log_diag[k * 64 + (j - 64)]);
        Wt[k * WTSTRIDE + j] = v;
    }
    if (tid < KK) la[tid] = log_alpha_raw[tid];   // EPSILON == 1
    __syncthreads();

    const int wave = tid >> 5;
    const int lane = tid & 31;
    float* xw = xbuf[wave];
    float* lw = lbuf[wave];
    int*   iw = idxbuf[wave];

    const int n    = lane & 15;          // N index within tile
    const int hsel = lane >> 4;          // 0 = lower half, 1 = upper half
    const int asel = hsel << 1;          // k-dim sub-select: 0 or 2
    const int half = hsel * 8;           // C-layout row offset

    // --- per-lane constant base pointers; all loop offsets are immediates ---
    const v2f* baseA2 = (const v2f*)(xw + n * XSTRIDE + asel);
    const v2f* baseB2 = (const v2f*)(Wt + n * WTSTRIDE + asel);
    float*     lstore = lw + half * XSTRIDE + n;           // logit scatter base
    const float* lscan = lw + (lane >> 1) * XSTRIDE + (lane & 1) * 32;
    const v2f* xo2    = (const v2f*)(xw + hsel * XSTRIDE + 4 * n);

    // bias hoisted out of the tile loop
    float lav0 = la[ 0 + n], lav1 = la[16 + n], lav2 = la[32 + n], lav3 = la[48 + n];

    for (int tile = blockIdx.x * WAVES + wave; tile < numTiles;
         tile += gridDim.x * WAVES) {

        const long rowbase = (long)tile * 16;
        const float* xt = x     + rowbase * DD;
        const float* ut = u     + rowbase * KK;
        const float* nt = noise + rowbase * DD;
        float*       ot = out   + rowbase * DD;

        // ---- stage x tile (coalesced global, padded LDS) ----
        #pragma unroll
        for (int i = 0; i < 8; ++i) {
            int e = i * 128 + lane * 4;          // e == row*64 + d
            int row = e >> 6, d = e & 63;
            float4 v = *(const float4*)(xt + e);
            v2f* dst = (v2f*)(xw + row * XSTRIDE + d);
            v2f lo, hi; lo[0] = v.x; lo[1] = v.y; hi[0] = v.z; hi[1] = v.w;
            dst[0] = lo; dst[1] = hi;
        }
        __builtin_amdgcn_wave_barrier();

        // ---- accumulators seeded with log_alpha (bias) ----
        v8f acc[4];
        #pragma unroll
        for (int m = 0; m < 8; ++m) {
            acc[0][m] = lav0; acc[1][m] = lav1; acc[2][m] = lav2; acc[3][m] = lav3;
        }

        // ---- fp32 WMMA GEMM, one-step software-pipelined fragment loads ----
        v2f a_cur = baseA2[0];
        v2f b0 = baseB2[0];
        v2f b1 = baseB2[ 8 * WTSTRIDE];
        v2f b2 = baseB2[16 * WTSTRIDE];
        v2f b3 = baseB2[24 * WTSTRIDE];
        #pragma unroll
        for (int kk = 0; kk < 32; ++kk) {
            v2f na = a_cur, nb0 = b0, nb1 = b1, nb2 = b2, nb3 = b3;
            if (kk < 31) {
                const int k2 = kk + 1;
                const int ai = (k2 < 16) ? 2 * k2 : 2 * (k2 - 16);
                na  = baseA2[ai];
                nb0 = baseB2[ 0 * WTSTRIDE + 2 * k2];
                nb1 = baseB2[ 8 * WTSTRIDE + 2 * k2];
                nb2 = baseB2[16 * WTSTRIDE + 2 * k2];
                nb3 = baseB2[24 * WTSTRIDE + 2 * k2];
            }
            v2f a = a_cur;
            if (kk >= 16) {                       // 0.5 * x^2 half of A
                a[0] = 0.5f * a[0] * a[0];
                a[1] = 0.5f * a[1] * a[1];
            }
            acc[0] = __builtin_amdgcn_wmma_f32_16x16x4_f32(false, a, false, b0, (short)0, acc[0], false, false);
            acc[1] = __builtin_amdgcn_wmma_f32_16x16x4_f32(false, a, false, b1, (short)0, acc[1], false, false);
            acc[2] = __builtin_amdgcn_wmma_f32_16x16x4_f32(false, a, false, b2, (short)0, acc[2], false, false);
            acc[3] = __builtin_amdgcn_wmma_f32_16x16x4_f32(false, a, false, b3, (short)0, acc[3], false, false);
            a_cur = na; b0 = nb0; b1 = nb1; b2 = nb2; b3 = nb3;
        }

        // ---- gumbel straight from global (C-layout), scatter logits to LDS --
        const float* ug = ut + half * 64 + n;     // + m*64 + t*16 (immediates)
        #pragma unroll
        for (int t = 0; t < 4; ++t) {
            #pragma unroll
            for (int m = 0; m < 8; ++m) {
                float uu = ug[m * 64 + t * 16];
                uu = fminf(fmaxf(uu, 1e-12f), 1.0f - 1e-7f);
                float l2 = __logf(-__logf(uu));
                lstore[m * XSTRIDE + t * 16] = acc[t][m] - l2;
            }
        }
        __builtin_amdgcn_wave_barrier();

        // ---- argmax over K=64: 2 lanes per row, 32 cols each ----
        {
            int c0 = (lane & 1) * 32;
            float best = lscan[0];
            int   bi   = c0;
            #pragma unroll
            for (int c = 1; c < 32; ++c) {
                float v = lscan[c];
                if (v > best) { best = v; bi = c0 + c; }
            }
            float ob = __shfl_xor(best, 1, 32);
            int   oi = __shfl_xor(bi, 1, 32);
            if (ob > best || (ob == best && oi < bi)) { best = ob; bi = oi; }
            if ((lane & 1) == 0) iw[lane >> 1] = bi;   // first-index tie-break
        }
        __builtin_amdgcn_wave_barrier();

        // ---- gathered output, fully coalesced loads/stores ----
        const v2f* w2 = (const v2f*)Wt;
        #pragma unroll
        for (int i = 0; i < 8; ++i) {
            int e = i * 128 + lane * 4;
            int d = (lane & 15) * 4;
            int k = iw[2 * i + hsel];
            v2f x01 = xo2[i * XSTRIDE];
            v2f x23 = xo2[i * XSTRIDE + 1];
            float4 nv = *(const float4*)(nt + e);
            int wb = (k * WTSTRIDE + d) >> 1;
            v2f r01 = w2[wb],      r23 = w2[wb + 1];
            v2f s01 = w2[wb + 32], s23 = w2[wb + 33];   // +64 floats = S block
            float4 ov;
            ov.x = fmaf(__builtin_amdgcn_sqrtf(s01[0]), nv.x, fmaf(s01[0], x01[0], r01[0]));
            ov.y = fmaf(__builtin_amdgcn_sqrtf(s01[1]), nv.y, fmaf(s01[1], x01[1], r01[1]));
            ov.z = fmaf(__builtin_amdgcn_sqrtf(s23[0]), nv.z, fmaf(s23[0], x23[0], r23[0]));
            ov.w = fmaf(__builtin_amdgcn_sqrtf(s23[1]), nv.w, fmaf(s23[1], x23[1], r23[1]));
            *(float4*)(ot + e) = ov;
        }
        __builtin_amdgcn_wave_barrier();   // WAR guard before next tile reuse
    }
}

// Scalar tail for B not divisible by 16 (unused for B=500000, kept for safety).
__global__ void lightsb_tail_kernel(const float* __restrict__ x,
                                    const float* __restrict__ log_alpha_raw,
                                    const float* __restrict__ r,
                                    const float* __restrict__ S_log_diag,
                                    const float* __restrict__ u,
                                    const float* __restrict__ noise,
                                    float* __restrict__ out,
                                    int rowStart, int Bn)
{
    int row = rowStart + blockIdx.x * blockDim.x + threadIdx.x;
    if (row >= Bn) return;
    float best = -INFINITY; int bi = 0;
    for (int k = 0; k < KK; ++k) {
        float acc = log_alpha_raw[k];
        for (int d = 0; d < DD; ++d) {
            float xv = x[row * DD + d];
            acc += xv * r[k * DD + d]
                 + 0.5f * xv * xv * expf(S_log_diag[k * DD + d]);
        }
        float uu = fminf(fmaxf(u[row * KK + k], 1e-12f), 1.0f - 1e-7f);
        float v = acc - logf(-logf(uu));
        if (v > best) { best = v; bi = k; }
    }
    for (int d = 0; d < DD; ++d) {
        float s = expf(S_log_diag[bi * DD + d]);
        out[row * DD + d] = r[bi * DD + d] + s * x[row * DD + d]
                          + sqrtf(s) * noise[row * DD + d];
    }
}

extern "C" void kernel_launch(void* const* d_in, const int* in_sizes, int n_in,
                              void* d_out, int out_size, void* d_ws, size_t ws_size,
                              hipStream_t stream) {
    const float* x     = (const float*)d_in[0];
    const float* lar   = (const float*)d_in[1];
    const float* r     = (const float*)d_in[2];
    const float* sld   = (const float*)d_in[3];
    const float* u     = (const float*)d_in[4];
    const float* noise = (const float*)d_in[5];
    float* out = (float*)d_out;

    const int Bn = in_sizes[0] / DD;          // 500000
    const int numTiles = Bn / 16;             // 31250 full 16-row tiles

    if (numTiles > 0) {
        int blocks = 1024;
        int need = (numTiles + WAVES - 1) / WAVES;
        if (blocks > need) blocks = need;
        lightsb_wmma_kernel<<<blocks, BLOCK, 0, stream>>>(
            x, lar, r, sld, u, noise, out, numTiles);
    }
    const int rem = Bn - numTiles * 16;
    if (rem > 0) {
        lightsb_tail_kernel<<<(rem + 63) / 64, 64, 0, stream>>>(
            x, lar, r, sld, u, noise, out, numTiles * 16, Bn);
    }
}